// MultiHeadISNEAttention_23622320128101
// MI455X (gfx1250) — compile-verified
//
#include <hip/hip_runtime.h>
#include <hip/hip_bf16.h>
#include <stdint.h>

#define ALPHA_SLOPE 0.2f
#define IN_F   64
#define OUT_F  512
#define N_HEADS 8
#define HEAD_DIM 64

typedef float v2f __attribute__((ext_vector_type(2)));
typedef float v8f __attribute__((ext_vector_type(8)));

// ---------------------------------------------------------------------------
// fp32 WMMA: D(16x16) = A(16x4) * B(4x16) + C   (V_WMMA_F32_16X16X4_F32)
// ---------------------------------------------------------------------------
__device__ __forceinline__ v8f wmma_f32(v2f a, v2f b, v8f c) {
  return __builtin_amdgcn_wmma_f32_16x16x4_f32(
      /*neg_a=*/false, a, /*neg_b=*/false, b,
      /*c_mod=*/(short)0, c, /*reuse_a=*/false, /*reuse_b=*/false);
}

// Monotonic float <-> uint mapping so unsigned atomicMax == float max.
__device__ __forceinline__ unsigned f2key(float f) {
  unsigned u = __float_as_uint(f);
  return (u & 0x80000000u) ? ~u : (u | 0x80000000u);
}
__device__ __forceinline__ float key2f(unsigned k) {
  return (k & 0x80000000u) ? __uint_as_float(k ^ 0x80000000u)
                           : __uint_as_float(~k);
}

// ---------------------------------------------------------------------------
// Zero-init kernel (accumulators live in poisoned workspace)
// ---------------------------------------------------------------------------
__global__ void zero_kernel(float* __restrict__ p, size_t count) {
  size_t i = (size_t)blockIdx.x * blockDim.x + threadIdx.x;
  if (i < count) p[i] = 0.0f;
}

// ---------------------------------------------------------------------------
// GEMM 1: Wh[n, h*64+d] = sum_i x[n,i] * W[h,i,d]
//   A = x row-major [n, 64];  B[i][c] = W[(c>>6)*4096 + i*64 + (c&63)]
// One wave per 16-row x 128-col strip (8 WMMA accumulators, A-frag reused).
// ---------------------------------------------------------------------------
__global__ __launch_bounds__(32) void gemm_wh_kernel(
    const float* __restrict__ x, const float* __restrict__ W,
    float* __restrict__ Wh, int n) {
  const int lane = threadIdx.x & 31;
  const int half = lane >> 4;      // 0: K{0,1} / M 0..7 ; 1: K{2,3} / M 8..15
  const int r    = lane & 15;
  const int row0 = blockIdx.x * 16;
  const int col0 = blockIdx.y * 128;

  v8f c[8];
#pragma unroll
  for (int t = 0; t < 8; ++t)
#pragma unroll
    for (int j = 0; j < 8; ++j) c[t][j] = 0.0f;

  int arow = row0 + r; if (arow >= n) arow = n - 1;

  for (int k0 = 0; k0 < IN_F; k0 += 4) {
    const int ka = k0 + 2 * half;
    v2f a = *(const v2f*)(x + arow * IN_F + ka);
#pragma unroll
    for (int t = 0; t < 8; ++t) {
      const int col = col0 + t * 16 + r;
      const int h = col >> 6, d = col & 63;
      v2f b;
      b.x = W[h * (IN_F * HEAD_DIM) + ka * HEAD_DIM + d];
      b.y = W[h * (IN_F * HEAD_DIM) + (ka + 1) * HEAD_DIM + d];
      c[t] = wmma_f32(a, b, c[t]);
    }
  }

  if (row0 + 16 <= n) {  // wave-uniform fast path: unguarded stores
#pragma unroll
    for (int t = 0; t < 8; ++t) {
      const int col = col0 + t * 16 + r;
#pragma unroll
      for (int j = 0; j < 8; ++j)
        Wh[(size_t)(row0 + j + 8 * half) * OUT_F + col] = c[t][j];
    }
  } else {
#pragma unroll
    for (int t = 0; t < 8; ++t) {
      const int col = col0 + t * 16 + r;
#pragma unroll
      for (int j = 0; j < 8; ++j) {
        const int row = row0 + j + 8 * half;
        if (row < n) Wh[(size_t)row * OUT_F + col] = c[t][j];
      }
    }
  }
}

// ---------------------------------------------------------------------------
// Per-(node, head) attention scores: s_src = Wh . a1, s_tgt = Wh . a2
// ---------------------------------------------------------------------------
__global__ void scores_kernel(const float* __restrict__ Wh,
                              const float* __restrict__ a1,
                              const float* __restrict__ a2,
                              float* __restrict__ s_src,
                              float* __restrict__ s_tgt, int n) {
  int idx = blockIdx.x * blockDim.x + threadIdx.x;
  if (idx >= n * N_HEADS) return;
  const int node = idx >> 3, h = idx & 7;
  const float4* w  = (const float4*)(Wh + (size_t)node * OUT_F + h * HEAD_DIM);
  const float4* v1 = (const float4*)(a1 + h * HEAD_DIM);
  const float4* v2 = (const float4*)(a2 + h * HEAD_DIM);
  float ss = 0.f, st = 0.f;
#pragma unroll
  for (int q = 0; q < HEAD_DIM / 4; ++q) {
    float4 wv = w[q], p1 = v1[q], p2 = v2[q];
    ss += wv.x * p1.x + wv.y * p1.y + wv.z * p1.z + wv.w * p1.w;
    st += wv.x * p2.x + wv.y * p2.y + wv.z * p2.z + wv.w * p2.w;
  }
  s_src[idx] = ss;
  s_tgt[idx] = st;
}

// ---------------------------------------------------------------------------
// Edge pass 1: segment max of leakyrelu(s_src[src] + s_tgt[tgt]) over src
// ---------------------------------------------------------------------------
__global__ void edge_max_kernel(const long long* __restrict__ src,
                                const long long* __restrict__ tgt,
                                const float* __restrict__ s_src,
                                const float* __restrict__ s_tgt,
                                unsigned* __restrict__ emax, int E) {
  int idx = blockIdx.x * blockDim.x + threadIdx.x;
  if (idx >= E * N_HEADS) return;
  const int e = idx >> 3, h = idx & 7;
  const long long s = src[e], t = tgt[e];
  float v = s_src[s * N_HEADS + h] + s_tgt[t * N_HEADS + h];
  v = (v > 0.f) ? v : ALPHA_SLOPE * v;
  atomicMax(&emax[s * N_HEADS + h], f2key(v));
}

// ---------------------------------------------------------------------------
// Edge pass 2: e_exp = exp(e - max), denom[src] += e_exp
// ---------------------------------------------------------------------------
__global__ void edge_exp_kernel(const long long* __restrict__ src,
                                const long long* __restrict__ tgt,
                                const float* __restrict__ s_src,
                                const float* __restrict__ s_tgt,
                                const unsigned* __restrict__ emax,
                                float* __restrict__ denom,
                                float* __restrict__ eexp, int E) {
  int idx = blockIdx.x * blockDim.x + threadIdx.x;
  if (idx >= E * N_HEADS) return;
  const int e = idx >> 3, h = idx & 7;
  const long long s = src[e], t = tgt[e];
  float v = s_src[s * N_HEADS + h] + s_tgt[t * N_HEADS + h];
  v = (v > 0.f) ? v : ALPHA_SLOPE * v;
  const float m = key2f(emax[s * N_HEADS + h]);
  const float ex = expf(v - m);
  eexp[idx] = ex;
  atomicAdd(&denom[s * N_HEADS + h], ex);
}

// ---------------------------------------------------------------------------
// Edge pass 3: head_out[src, h, :] += (e_exp/denom[src]) * Wh[tgt, h, :]
// 32 consecutive threads cover the 64-wide d-dimension of one (edge, head):
// fully coalesced 128B loads from Wh[tgt] and coalesced f32 atomics.
// ---------------------------------------------------------------------------
__global__ void edge_scatter_kernel(const long long* __restrict__ src,
                                    const long long* __restrict__ tgt,
                                    const float* __restrict__ eexp,
                                    const float* __restrict__ denom,
                                    const float* __restrict__ Wh,
                                    float* __restrict__ head_out, int E) {
  const int gid = blockIdx.x * blockDim.x + threadIdx.x;
  const int eh  = gid >> 5;          // (edge, head) pair
  const int d   = gid & 31;          // lane within d-dimension
  if (eh >= E * N_HEADS) return;
  const int e = eh >> 3, h = eh & 7;
  const long long s = src[e], t = tgt[e];
  const float att = eexp[eh] / denom[s * N_HEADS + h];
  const float* wv = Wh + (size_t)t * OUT_F + h * HEAD_DIM;
  float* ho = head_out + (size_t)s * OUT_F + h * HEAD_DIM;
  atomicAdd(ho + d,      att * wv[d]);
  atomicAdd(ho + d + 32, att * wv[d + 32]);
}

// ---------------------------------------------------------------------------
// GEMM 3 (fused): out = head_out @ Pᵀ + x @ Rᵀ + pb + rb
//   B1[k][o] = P[o*512 + k];  B2[i][o] = R[o*64 + i]
// One wave per 16-row x 128-col strip.
// ---------------------------------------------------------------------------
__global__ __launch_bounds__(32) void gemm_out_kernel(
    const float* __restrict__ head, const float* __restrict__ x,
    const float* __restrict__ P, const float* __restrict__ pb,
    const float* __restrict__ R, const float* __restrict__ rb,
    float* __restrict__ out, int n) {
  const int lane = threadIdx.x & 31;
  const int half = lane >> 4;
  const int r    = lane & 15;
  const int row0 = blockIdx.x * 16;
  const int col0 = blockIdx.y * 128;

  // Warm bias vectors into cache while the K-loop runs (global_prefetch_b8).
  __builtin_prefetch(pb + col0 + r, 0, 3);
  __builtin_prefetch(rb + col0 + r, 0, 3);

  v8f c[8];
#pragma unroll
  for (int t = 0; t < 8; ++t)
#pragma unroll
    for (int j = 0; j < 8; ++j) c[t][j] = 0.0f;

  int arow = row0 + r; if (arow >= n) arow = n - 1;

  // main term: K = 512 over head_out
  for (int k0 = 0; k0 < OUT_F; k0 += 4) {
    const int ka = k0 + 2 * half;
    v2f a = *(const v2f*)(head + (size_t)arow * OUT_F + ka);
#pragma unroll
    for (int t = 0; t < 8; ++t) {
      const int col = col0 + t * 16 + r;
      v2f b = *(const v2f*)(P + (size_t)col * OUT_F + ka);
      c[t] = wmma_f32(a, b, c[t]);
    }
  }

  // residual term: K = 64 over x
  for (int k0 = 0; k0 < IN_F; k0 += 4) {
    const int ka = k0 + 2 * half;
    v2f a = *(const v2f*)(x + arow * IN_F + ka);
#pragma unroll
    for (int t = 0; t < 8; ++t) {
      const int col = col0 + t * 16 + r;
      v2f b = *(const v2f*)(R + col * IN_F + ka);
      c[t] = wmma_f32(a, b, c[t]);
    }
  }

  if (row0 + 16 <= n) {  // wave-uniform fast path
#pragma unroll
    for (int t = 0; t < 8; ++t) {
      const int col = col0 + t * 16 + r;
      const float bias = pb[col] + rb[col];
#pragma unroll
      for (int j = 0; j < 8; ++j)
        out[(size_t)(row0 + j + 8 * half) * OUT_F + col] = c[t][j] + bias;
    }
  } else {
#pragma unroll
    for (int t = 0; t < 8; ++t) {
      const int col = col0 + t * 16 + r;
      const float bias = pb[col] + rb[col];
#pragma unroll
      for (int j = 0; j < 8; ++j) {
        const int row = row0 + j + 8 * half;
        if (row < n) out[(size_t)row * OUT_F + col] = c[t][j] + bias;
      }
    }
  }
}

// ---------------------------------------------------------------------------
// Host-side launcher
// ---------------------------------------------------------------------------
extern "C" void kernel_launch(void* const* d_in, const int* in_sizes, int n_in,
                              void* d_out, int out_size, void* d_ws, size_t ws_size,
                              hipStream_t stream) {
  const float*     x    = (const float*)d_in[0];
  const long long* eidx = (const long long*)d_in[1];  // int64 [2, E]
  const float*     W    = (const float*)d_in[2];      // [8, 64, 64]
  const float*     a1   = (const float*)d_in[3];      // [8, 64]
  const float*     a2   = (const float*)d_in[4];      // [8, 64]
  const float*     opw  = (const float*)d_in[5];      // [512, 512]
  const float*     opb  = (const float*)d_in[6];      // [512]
  const float*     rw   = (const float*)d_in[7];      // [512, 64]
  const float*     rb   = (const float*)d_in[8];      // [512]
  float* out = (float*)d_out;

  const int n = in_sizes[0] / IN_F;   // 50000
  const int E = in_sizes[1] / 2;      // 400000
  const long long* src = eidx;
  const long long* tgt = eidx + E;

  // Workspace layout (floats). Accumulators (head_out, emax, denom) contiguous
  // so one zero-init kernel covers them.
  float* ws = (float*)d_ws;
  float*    Wh       = ws;                                        // n*512
  float*    head_out = Wh + (size_t)n * OUT_F;                    // n*512
  unsigned* emax     = (unsigned*)(head_out + (size_t)n * OUT_F); // n*8
  float*    denom    = (float*)emax + (size_t)n * N_HEADS;        // n*8
  float*    s_src    = denom + (size_t)n * N_HEADS;               // n*8
  float*    s_tgt    = s_src + (size_t)n * N_HEADS;               // n*8
  float*    eexp     = s_tgt + (size_t)n * N_HEADS;               // E*8

  const size_t zcount = (size_t)n * OUT_F + 2 * (size_t)n * N_HEADS;
  zero_kernel<<<(unsigned)((zcount + 255) / 256), 256, 0, stream>>>(head_out, zcount);

  dim3 gemm_grid((n + 15) / 16, OUT_F / 128);
  gemm_wh_kernel<<<gemm_grid, 32, 0, stream>>>(x, W, Wh, n);

  const int nh = n * N_HEADS;
  scores_kernel<<<(nh + 255) / 256, 256, 0, stream>>>(Wh, a1, a2, s_src, s_tgt, n);

  const int EH = E * N_HEADS;
  edge_max_kernel<<<(EH + 255) / 256, 256, 0, stream>>>(src, tgt, s_src, s_tgt, emax, E);
  edge_exp_kernel<<<(EH + 255) / 256, 256, 0, stream>>>(src, tgt, s_src, s_tgt, emax, denom, eexp, E);

  const long long scat_threads = (long long)EH * 32;
  edge_scatter_kernel<<<(unsigned)((scat_threads + 255) / 256), 256, 0, stream>>>(
      src, tgt, eexp, denom, Wh, head_out, E);

  gemm_out_kernel<<<gemm_grid, 32, 0, stream>>>(head_out, x, opw, opb, rw, rb, out, n);
}